// MaxUnpooling2D_1082331758744
// MI455X (gfx1250) — compile-verified
//
#include <hip/hip_runtime.h>

// MaxUnpooling2D for B=8, H=128, W=128, C=64, pool 2x2 -> out (8,256,256,64).
//
// Pure scatter, zero FLOPs: bandwidth-roofline kernel for MI455X (23.3 TB/s).
// Strategy: gather-ify the scatter. Each thread owns 4 consecutive channels of
// one pooled pixel and writes ALL FOUR 2x2 output slots (value or 0), so:
//   - no separate 128 MiB zero-fill pass (every output byte written once)
//   - all loads/stores are b128, coalesced across the wave32
//   - non-temporal (TH=NT) cache hints: streaming data, touched exactly once
// Total traffic = 32 MiB (upd) + 32 MiB (mask) + 128 MiB (out) ~= 8.6 us peak.

typedef __attribute__((ext_vector_type(4))) float v4f;
typedef __attribute__((ext_vector_type(4))) int   v4i;

// Compile-time geometry (matches the reference's fixed shapes).
// in element i = ((b*H + y)*W + x)*C + f ; i = 4*t for thread t.
//   b = i >> 20            (H*W*C = 2^20)
//   outVol = 256*256*64 = 2^22 floats per batch
// mask m = (y_out*256 + x_out)*64 + f:
//   ox = (m>>6)&1, oy = (m>>14)&1  (no carries: f<64, x_out<256)
//   top-left slot base = m & ~((1<<14)|(1<<6)) = m & ~0x4040
// slot offsets in floats: x+1 -> +64, y+1 -> +16384.

__global__ __launch_bounds__(256) void max_unpool2d_scatter(
    const v4f* __restrict__ upd,
    const v4i* __restrict__ msk,
    float*     __restrict__ out,
    int n4)
{
    int t = blockIdx.x * 256 + threadIdx.x;     // group-of-4-channels index
    if (t >= n4) return;

    v4i m = __builtin_nontemporal_load(msk + t);   // global_load_b128 th:NT
    v4f u = __builtin_nontemporal_load(upd + t);   // global_load_b128 th:NT

    int b    = t >> 18;                          // (4t) / 2^20
    int base = (b << 22) + (m.x & ~0x4040);      // batch base + top-left slot
                                                 // (f of comp0 is mult of 4 ->
                                                 //  16B-aligned b128 stores)

    // 2x2 slot id per channel: slot = oy*2 + ox
    int s0 = ((m.x >> 13) & 2) | ((m.x >> 6) & 1);
    int s1 = ((m.y >> 13) & 2) | ((m.y >> 6) & 1);
    int s2 = ((m.z >> 13) & 2) | ((m.z >> 6) & 1);
    int s3 = ((m.w >> 13) & 2) | ((m.w >> 6) & 1);

    float* p = out + base;
#pragma unroll
    for (int j = 0; j < 4; ++j) {
        v4f o;
        o.x = (s0 == j) ? u.x : 0.0f;            // v_cndmask_b32
        o.y = (s1 == j) ? u.y : 0.0f;
        o.z = (s2 == j) ? u.z : 0.0f;
        o.w = (s3 == j) ? u.w : 0.0f;
        int off = ((j & 1) << 6) | ((j >> 1) << 14);   // 0, 64, 16384, 16448
        __builtin_nontemporal_store(o, (v4f*)(p + off)); // global_store_b128 th:NT
    }
}

extern "C" void kernel_launch(void* const* d_in, const int* in_sizes, int n_in,
                              void* d_out, int out_size, void* d_ws, size_t ws_size,
                              hipStream_t stream) {
    const v4f* upd = (const v4f*)d_in[0];   // updates: (8,128,128,64) f32
    const v4i* msk = (const v4i*)d_in[1];   // mask:    (8,128,128,64) i32
    float*     out = (float*)d_out;         // out:     (8,256,256,64) f32

    int n  = in_sizes[0];                   // 8*128*128*64 = 8,388,608
    int n4 = n >> 2;                        // 2,097,152 threads
    int blocks = (n4 + 255) / 256;          // 8192 blocks = 65,536 wave32s

    max_unpool2d_scatter<<<blocks, 256, 0, stream>>>(upd, msk, out, n4);
}